// CTCLoss_43568148250855
// MI455X (gfx1250) — compile-verified
//
#include <hip/hip_runtime.h>

#define T_STEPS 2048
#define BATCH   64
#define VOCAB   128
#define LMAX_   128
#define S_EXT   257           // 2*LMAX+1
#define NEGF    (-1e30f)

// ---------------------------------------------------------------------------
// Kernel A (placed first so its asm leads the disasm): serial alpha recursion.
// One block per batch element b. 288 threads (9 wave32s): threads 0..256 each
// own one lattice state. acts rows stream into a 4-slot LDS ring via CDNA5
// async-tensor loads (global_load_async_to_lds_b128, one wave x 16B/lane =
// one 512B row per issue), prefetch distance 2, waited with s_wait_asynccnt.
// Exactly one s_barrier per time step; branch-free steady-state loop.
// ---------------------------------------------------------------------------
__global__ __launch_bounds__(288) void ctc_alpha_kernel(
    const float* __restrict__ acts, const int* __restrict__ labels,
    const int* __restrict__ act_lens, const int* __restrict__ label_lens,
    const float* __restrict__ lse, float* __restrict__ costs) {
  __shared__ float abuf[4 * VOCAB];          // 4-slot ring of acts rows (2 KiB)
  __shared__ float alpha[2][S_EXT + 7];      // double-buffered lattice
  __shared__ float lse_sm[T_STEPS];          // staged logsumexp column (8 KiB)
  __shared__ int   s_off;

  const int b   = blockIdx.x;
  const int tid = threadIdx.x;
  const int len = act_lens[b];               // early-exit bound (freeze semantics)
  const int Lb  = label_lens[b];

  if (tid == 0) {                            // label offset = exclusive prefix sum
    int off = 0;
    for (int j = 0; j < b; ++j) off += label_lens[j];
    s_off = off;
  }
  for (int i = tid; i < T_STEPS; i += 288)   // stage lse column
    lse_sm[i] = lse[(size_t)b * T_STEPS + i];
  __syncthreads();
  const int off_b = s_off;

  // Per-thread lattice metadata: ext symbol + skip-transition flag.
  int  ext_s  = 0;
  bool skip_s = false;
  if (tid < S_EXT) {
    if (tid & 1) {
      const int j = (tid - 1) >> 1;
      ext_s = (j < Lb) ? labels[off_b + j] : 0;
      const int labm2 = (j == 0) ? -1 : ((j - 1 < Lb) ? labels[off_b + j - 1] : 0);
      skip_s = (ext_s != 0) && (ext_s != labm2);
    }
    alpha[0][tid] = (tid == 0) ? 0.0f : NEGF;
  }

  // Async-load machinery (wave 0 only issues).
  const unsigned lds_base = (unsigned)(unsigned long long)(&abuf[0]);
  const unsigned long long gbase =
      (unsigned long long)(acts + (size_t)b * VOCAB);    // row stride = B*V*4 = 32768B

#define ASYNC_LOAD_ROW(t_)                                                        \
  do {                                                                            \
    unsigned ldsa = lds_base + (((unsigned)(t_)&3u) * (VOCAB * 4)) + (lane << 4); \
    unsigned voff = ((unsigned)(t_) << 15) + (lane << 4);                         \
    asm volatile("global_load_async_to_lds_b128 %0, %1, %2"                       \
                 :: "v"(ldsa), "v"(voff), "s"(gbase) : "memory");                 \
  } while (0)

  if (tid < 32) {
    const int lane = tid;
    ASYNC_LOAD_ROW(0);
    if (len > 1) ASYNC_LOAD_ROW(1);
  }

  // One DP step: read row t from the LDS ring, update all 257 states.
  auto step = [&](int t) {
    const float* ab = &abuf[(t & 3) * VOCAB];
    const float emit = ab[ext_s] - lse_sm[t];
    const float* ac = alpha[t & 1];
    const float a0 = ac[tid];
    const float a1 = (tid >= 1) ? ac[tid - 1] : NEGF;
    const float a2 = (tid >= 2 && skip_s) ? ac[tid - 2] : NEGF;
    const float m  = fmaxf(a0, fmaxf(a1, a2));
    const float r  = __expf(a0 - m) + __expf(a1 - m) + __expf(a2 - m);
    alpha[(t + 1) & 1][tid] = emit + m + __logf(r);
  };

  int t = 0;
  const int steady = len - 2;                // iterations that prefetch t+2
  for (; t < steady; ++t) {                  // branch-free hot loop
    if (tid < 32) {
      const int lane = tid;
      ASYNC_LOAD_ROW(t + 2);
      asm volatile("s_wait_asynccnt 0x2" ::: "memory");  // row t retired (in-order)
    }
    __syncthreads();                         // row t + alpha buffers visible
    if (tid < S_EXT) step(t);
  }
  for (; t < len; ++t) {                     // <=2 tail iterations
    if (tid < 32) {
      if (t + 1 < len) asm volatile("s_wait_asynccnt 0x1" ::: "memory");
      else             asm volatile("s_wait_asynccnt 0x0" ::: "memory");
    }
    __syncthreads();
    if (tid < S_EXT) step(t);
  }
#undef ASYNC_LOAD_ROW

  __syncthreads();
  if (tid == 0) {
    const float* af = alpha[len & 1];
    const float x = af[2 * Lb];
    const float y = af[2 * Lb - 1];
    const float m = fmaxf(x, y);
    costs[b] = -(m + __logf(__expf(x - m) + __expf(y - m)));
  }
}

// ---------------------------------------------------------------------------
// Kernel B: per-(t,b) logsumexp over V=128. One wave32 per row, float4 loads
// (16B/lane), butterfly reductions. Fully parallel: 131072 rows; memory-bound
// (64 MiB @ 23.3 TB/s ~ 3us). Output layout lse[b*T + t] so kernel A stages a
// contiguous column.
// ---------------------------------------------------------------------------
__global__ __launch_bounds__(256) void ctc_lse_kernel(const float* __restrict__ acts,
                                                      float* __restrict__ lse) {
  const int wave = threadIdx.x >> 5;
  const int lane = threadIdx.x & 31;
  const int row  = blockIdx.x * 8 + wave;                 // row = t*BATCH + b
  const float4* p = reinterpret_cast<const float4*>(acts) + (size_t)row * (VOCAB / 4);
  float4 v = p[lane];

  float m = fmaxf(fmaxf(v.x, v.y), fmaxf(v.z, v.w));
#pragma unroll
  for (int off = 16; off >= 1; off >>= 1)
    m = fmaxf(m, __shfl_xor(m, off, 32));

  float s = __expf(v.x - m) + __expf(v.y - m) + __expf(v.z - m) + __expf(v.w - m);
#pragma unroll
  for (int off = 16; off >= 1; off >>= 1)
    s += __shfl_xor(s, off, 32);

  if (lane == 0) {
    const int t = row / BATCH;
    const int b = row - t * BATCH;
    lse[(size_t)b * T_STEPS + t] = m + __logf(s);
  }
}

// ---------------------------------------------------------------------------
// Kernel C: deterministic fixed-order sum of the 64 per-batch costs.
// ---------------------------------------------------------------------------
__global__ void ctc_sum_kernel(const float* __restrict__ costs, float* __restrict__ out) {
  __shared__ float sm[BATCH];
  sm[threadIdx.x] = costs[threadIdx.x];
  __syncthreads();
  if (threadIdx.x == 0) {
    float s = 0.0f;
    for (int i = 0; i < BATCH; ++i) s += sm[i];
    out[0] = s;
  }
}

extern "C" void kernel_launch(void* const* d_in, const int* in_sizes, int n_in,
                              void* d_out, int out_size, void* d_ws, size_t ws_size,
                              hipStream_t stream) {
  (void)in_sizes; (void)n_in; (void)out_size; (void)ws_size;
  const float* acts       = (const float*)d_in[0];
  const int*   labels     = (const int*)d_in[1];
  const int*   act_lens   = (const int*)d_in[2];
  const int*   label_lens = (const int*)d_in[3];
  float*       out        = (float*)d_out;

  float* lse   = (float*)d_ws;                        // T*B floats = 512 KiB
  float* costs = lse + (size_t)T_STEPS * BATCH;       // 64 floats

  ctc_lse_kernel<<<(T_STEPS * BATCH) / 8, 256, 0, stream>>>(acts, lse);
  ctc_alpha_kernel<<<BATCH, 288, 0, stream>>>(acts, labels, act_lens, label_lens,
                                              lse, costs);
  ctc_sum_kernel<<<1, BATCH, 0, stream>>>(costs, out);
}